// GCN_5342939316741
// MI455X (gfx1250) — compile-verified
//
#include <hip/hip_runtime.h>

// GCN 2-layer for MI455X (gfx1250, wave32).
// Memory-bound in the edge scatter (≈1.1 GB, L2-resident since h = 25.6 MB << 192 MB L2);
// GEMMs done with fp32 WMMA (V_WMMA_F32_16X16X4_F32) to keep exact fp32 semantics.

typedef float v2f __attribute__((ext_vector_type(2)));
typedef float v8f __attribute__((ext_vector_type(8)));

#define WMMA_F32X4(a, b, c) \
  __builtin_amdgcn_wmma_f32_16x16x4_f32(false, (a), false, (b), (short)0, (c), false, false)

// ---------------- degree / norm ----------------
__global__ void k_deg_init(float* deg, int n) {
  int i = blockIdx.x * blockDim.x + threadIdx.x;
  if (i < n) deg[i] = 1.0f;  // self-loop
}

__global__ void k_deg_edges(const long long* __restrict__ ei, float* deg, long long E) {
  long long e = (long long)blockIdx.x * blockDim.x + threadIdx.x;
  if (e < E) {
    int dst = (int)ei[E + e];
    unsafeAtomicAdd(&deg[dst], 1.0f);  // global_atomic_add_f32
  }
}

__global__ void k_rsqrt(float* dinv, int n) {
  int i = blockIdx.x * blockDim.x + threadIdx.x;
  if (i < n) dinv[i] = rsqrtf(dinv[i]);
}

// ---------------- GEMM1: h = x @ W1  (N x 64) @ (64 x 64) ----------------
// 8 waves / block; each wave computes a 16-row x 64-col tile with 64 WMMAs.
__global__ __launch_bounds__(256) void k_gemm1(const float* __restrict__ x,
                                               const float* __restrict__ W1,
                                               float* __restrict__ h,
                                               int nrb, int n) {
  __shared__ float sW[64 * 64];
  for (int i = threadIdx.x; i < 64 * 64; i += 256) sW[i] = W1[i];
  __syncthreads();

  const int wave = threadIdx.x >> 5;
  const int lane = threadIdx.x & 31;
  const int rb = blockIdx.x * 8 + wave;
  if (rb >= nrb) return;  // wave-uniform: EXEC stays all-1s for WMMA

  const int hi   = lane >> 4;        // 0 or 1 (half-wave)
  const int nn   = lane & 15;        // N index / M index within half
  const int koff = hi << 1;          // K sub-offset {0,2}
  int row = rb * 16 + nn;
  if (row >= n) row = n - 1;         // clamp tail reads (uniform-safe: no divergence in EXEC)
  const float* xrow = x + (long long)row * 64;

  v8f a0 = {}, a1 = {}, a2 = {}, a3 = {};
  for (int k0 = 0; k0 < 64; k0 += 4) {
    v2f a;
    a.x = xrow[k0 + koff];
    a.y = xrow[k0 + koff + 1];
    const float* w0 = sW + (k0 + koff) * 64;
    const float* w1 = sW + (k0 + koff + 1) * 64;
    v2f b0; b0.x = w0[nn];      b0.y = w1[nn];
    v2f b1; b1.x = w0[16 + nn]; b1.y = w1[16 + nn];
    v2f b2; b2.x = w0[32 + nn]; b2.y = w1[32 + nn];
    v2f b3; b3.x = w0[48 + nn]; b3.y = w1[48 + nn];
    a0 = WMMA_F32X4(a, b0, a0);
    a1 = WMMA_F32X4(a, b1, a1);
    a2 = WMMA_F32X4(a, b2, a2);
    a3 = WMMA_F32X4(a, b3, a3);
  }

  const int rbase = rb * 16 + (hi << 3);  // C/D: VGPR i -> M = i + 8*hi
#pragma unroll
  for (int i = 0; i < 8; ++i) {
    float* hr = h + (long long)(rbase + i) * 64 + nn;
    hr[0]  = a0[i];
    hr[16] = a1[i];
    hr[32] = a2[i];
    hr[48] = a3[i];
  }
}

// ---------------- layer-1 aggregation ----------------
__global__ void k_init1(const float* __restrict__ h, const float* __restrict__ dinv,
                        float* __restrict__ agg1, int n) {
  int gid = blockIdx.x * blockDim.x + threadIdx.x;  // n*16 threads, float4 each
  if (gid >= n * 16) return;
  int node = gid >> 4;
  int c4 = (gid & 15) << 2;
  float s = dinv[node] * dinv[node];
  const float4 v = *(const float4*)(h + (long long)node * 64 + c4);
  float4 o = make_float4(v.x * s, v.y * s, v.z * s, v.w * s);
  *(float4*)(agg1 + (long long)node * 64 + c4) = o;
}

__global__ void k_scat1(const long long* __restrict__ ei, const float* __restrict__ h,
                        const float* __restrict__ dinv, float* __restrict__ agg1,
                        long long E) {
  long long gid = (long long)blockIdx.x * blockDim.x + threadIdx.x;  // E*16 threads
  if (gid >= E * 16) return;
  long long e = gid >> 4;
  int c4 = ((int)gid & 15) << 2;
  int src = (int)ei[e];
  int dst = (int)ei[E + e];
  float w = dinv[src] * dinv[dst];
  const float4 v = *(const float4*)(h + (long long)src * 64 + c4);
  float* p = agg1 + (long long)dst * 64 + c4;
  unsafeAtomicAdd(p + 0, v.x * w);
  unsafeAtomicAdd(p + 1, v.y * w);
  unsafeAtomicAdd(p + 2, v.z * w);
  unsafeAtomicAdd(p + 3, v.w * w);
}

// ---------------- GEMM2: h2 = relu(agg1 + b1) @ W2  (N x 64) @ (64 x 16) ----------------
__global__ __launch_bounds__(256) void k_gemm2(const float* __restrict__ agg1,
                                               const float* __restrict__ b1,
                                               const float* __restrict__ W2,
                                               float* __restrict__ h2,
                                               int nrb, int n) {
  __shared__ float sW[64 * 16];
  __shared__ float sb[64];
  for (int i = threadIdx.x; i < 64 * 16; i += 256) sW[i] = W2[i];
  if (threadIdx.x < 64) sb[threadIdx.x] = b1[threadIdx.x];
  __syncthreads();

  const int wave = threadIdx.x >> 5;
  const int lane = threadIdx.x & 31;
  const int rb = blockIdx.x * 8 + wave;
  if (rb >= nrb) return;

  const int hi   = lane >> 4;
  const int nn   = lane & 15;
  const int koff = hi << 1;
  int row = rb * 16 + nn;
  if (row >= n) row = n - 1;
  const float* arow = agg1 + (long long)row * 64;

  v8f acc = {};
  for (int k0 = 0; k0 < 64; k0 += 4) {
    v2f a;
    a.x = fmaxf(arow[k0 + koff] + sb[k0 + koff], 0.0f);          // fused bias + relu
    a.y = fmaxf(arow[k0 + koff + 1] + sb[k0 + koff + 1], 0.0f);
    v2f b;
    b.x = sW[(k0 + koff) * 16 + nn];
    b.y = sW[(k0 + koff + 1) * 16 + nn];
    acc = WMMA_F32X4(a, b, acc);
  }

  const int rbase = rb * 16 + (hi << 3);
#pragma unroll
  for (int i = 0; i < 8; ++i)
    h2[(long long)(rbase + i) * 16 + nn] = acc[i];
}

// ---------------- layer-2 aggregation ----------------
__global__ void k_init2(const float* __restrict__ h2, const float* __restrict__ dinv,
                        const float* __restrict__ b2, float* __restrict__ out, int n) {
  int gid = blockIdx.x * blockDim.x + threadIdx.x;  // n*16 threads
  if (gid >= n * 16) return;
  int node = gid >> 4;
  int c = gid & 15;
  float s = dinv[node] * dinv[node];
  out[gid] = h2[gid] * s + b2[c];
}

__global__ void k_scat2(const long long* __restrict__ ei, const float* __restrict__ h2,
                        const float* __restrict__ dinv, float* __restrict__ out,
                        long long E) {
  long long gid = (long long)blockIdx.x * blockDim.x + threadIdx.x;  // E*16 threads
  if (gid >= E * 16) return;
  long long e = gid >> 4;
  int c = (int)gid & 15;
  int src = (int)ei[e];
  int dst = (int)ei[E + e];
  float w = dinv[src] * dinv[dst];
  unsafeAtomicAdd(out + (long long)dst * 16 + c, h2[(long long)src * 16 + c] * w);
}

extern "C" void kernel_launch(void* const* d_in, const int* in_sizes, int n_in,
                              void* d_out, int out_size, void* d_ws, size_t ws_size,
                              hipStream_t stream) {
  const float* x      = (const float*)d_in[0];
  const long long* ei = (const long long*)d_in[1];  // int64 edge_index, shape (2, E)
  const float* W1     = (const float*)d_in[2];
  const float* b1     = (const float*)d_in[3];
  const float* W2     = (const float*)d_in[4];
  const float* b2     = (const float*)d_in[5];
  float* out          = (float*)d_out;

  const int N = in_sizes[0] / 64;
  const long long E = in_sizes[1] / 2;
  const int nrb = (N + 15) / 16;
  const int Np = nrb * 16;

  // workspace layout (floats): dinv[Np] | h[Np*64] (h2 aliases first Np*16) | agg1[Np*64]
  float* dinv = (float*)d_ws;
  float* h    = dinv + Np;
  float* agg1 = h + (size_t)Np * 64;
  float* h2   = h;  // h is dead after k_scat1; reuse its space for h2

  const int T = 256;
  // degrees -> dinv
  k_deg_init<<<(N + T - 1) / T, T, 0, stream>>>(dinv, N);
  k_deg_edges<<<(unsigned)((E + T - 1) / T), T, 0, stream>>>(ei, dinv, E);
  k_rsqrt<<<(N + T - 1) / T, T, 0, stream>>>(dinv, N);
  // layer 1
  k_gemm1<<<(nrb + 7) / 8, T, 0, stream>>>(x, W1, h, nrb, N);
  k_init1<<<(N * 16 + T - 1) / T, T, 0, stream>>>(h, dinv, agg1, N);
  k_scat1<<<(unsigned)((E * 16 + T - 1) / T), T, 0, stream>>>(ei, h, dinv, agg1, E);
  // layer 2 (bias+relu fused into GEMM2 A-operand)
  k_gemm2<<<(nrb + 7) / 8, T, 0, stream>>>(agg1, b1, W2, h2, nrb, N);
  k_init2<<<(N * 16 + T - 1) / T, T, 0, stream>>>(h2, dinv, b2, out, N);
  k_scat2<<<(unsigned)((E * 16 + T - 1) / T), T, 0, stream>>>(ei, h2, dinv, out, E);
}